// GraphSAGEModel_24326694764904
// MI455X (gfx1250) — compile-verified
//
#include <hip/hip_runtime.h>

// GraphSAGE 3-layer forward for MI455X (gfx1250).
// N=100000 nodes, E=1600000 edges, all dims 64.
// Aggregation: f32 atomics into L2-resident msum (25.6 MB << 192 MB L2).
// GEMM: V_WMMA_F32_16X16X4_F32, one 16x16 tile per wave32.

#define N_NODES 100000
#define N_EDGES 1600000
#define F 64
#define HSTRIDE 65  // LDS row stride padding (avoid 16-way bank conflicts)

typedef __attribute__((ext_vector_type(2))) float v2f;
typedef __attribute__((ext_vector_type(8))) float v8f;

__global__ __launch_bounds__(256) void zero_f32(float* __restrict__ p, long long n) {
    long long i = (long long)blockIdx.x * 256 + threadIdx.x;
    long long stride = (long long)gridDim.x * 256;
    for (; i < n; i += stride) p[i] = 0.0f;
}

__global__ __launch_bounds__(256) void sage_degree(const int* __restrict__ dst,
                                                   float* __restrict__ deg, int n_edges) {
    int e = blockIdx.x * 256 + threadIdx.x;
    if (e < n_edges) atomicAdd(&deg[dst[e]], 1.0f);
}

// One edge per 32 lanes; each lane moves 2 consecutive floats (float2 load, 2 f32 atomics).
__global__ __launch_bounds__(256) void sage_aggregate(const float* __restrict__ h,
                                                      const int* __restrict__ src,
                                                      const int* __restrict__ dst,
                                                      float* __restrict__ msum, int n_edges) {
    long long idx = (long long)blockIdx.x * 256 + threadIdx.x;
    int e = (int)(idx >> 5);
    int l = (int)(idx & 31);
    if (e >= n_edges) return;
    int s = src[e];
    int d = dst[e];
    const float2 v = *(const float2*)(h + (long long)s * F + l * 2);
    atomicAdd(&msum[(long long)d * F + l * 2 + 0], v.x);
    atomicAdd(&msum[(long long)d * F + l * 2 + 1], v.y);
}

// Block: 256 threads = 8 waves, 32 nodes. Each wave computes one 16x16 output tile
// via 32x V_WMMA_F32_16X16X4_F32 (16 K-steps of 4, self + neigh fused per step).
__global__ __launch_bounds__(256) void sage_gemm(const float* __restrict__ h,
                                                 const float* __restrict__ msum,
                                                 const float* __restrict__ deg,
                                                 const float* __restrict__ Wself,
                                                 const float* __restrict__ Wneigh,
                                                 const float* __restrict__ bias,
                                                 float* __restrict__ out,
                                                 int n_nodes, int do_relu) {
    __shared__ float sWs[F * F];          // 16 KB, row-major [k][n]
    __shared__ float sWn[F * F];          // 16 KB
    __shared__ float sH [32 * HSTRIDE];   // ~8.1 KB, padded rows
    __shared__ float sHn[32 * HSTRIDE];   // ~8.1 KB
    __shared__ float sB [F];

    const int tid = threadIdx.x;
    const int node0 = blockIdx.x * 32;

    // Stage both weight matrices (coalesced, 16 floats/thread each).
    #pragma unroll
    for (int i = 0; i < 16; ++i) {
        sWs[tid + i * 256] = Wself [tid + i * 256];
        sWn[tid + i * 256] = Wneigh[tid + i * 256];
    }
    if (tid < F) sB[tid] = bias[tid];

    // Stage h block and h_neigh = msum * (1/max(deg,1)) (8 floats/thread).
    #pragma unroll
    for (int i = 0; i < 8; ++i) {
        int idx = tid + i * 256;     // 0..2047
        int row = idx >> 6;          // 0..31
        int col = idx & 63;
        int node = node0 + row;
        float hv = 0.0f, mv = 0.0f, dv = 1.0f;
        if (node < n_nodes) {
            hv = h   [(long long)node * F + col];
            mv = msum[(long long)node * F + col];
            dv = deg[node];
        }
        float rinv = 1.0f / fmaxf(dv, 1.0f);
        sH [row * HSTRIDE + col] = hv;
        sHn[row * HSTRIDE + col] = mv * rinv;
    }
    __syncthreads();

    const int wave  = tid >> 5;      // 0..7
    const int lane  = tid & 31;
    const int m0    = (wave >> 2) * 16;  // row tile within block: 0 or 16
    const int c0    = (wave & 3) * 16;   // col tile: 0,16,32,48
    const int lrow  = lane & 15;         // 0..15
    const int khalf = lane >> 4;         // 0 or 1

    // A 16x4 f32 layout: lanes 0-15 hold M=0..15 / K={0,1}; lanes 16-31 K={2,3}.
    // B 4x16 mirrored: lanes 0-15 hold N=0..15 of rows K={0,1}; lanes 16-31 K={2,3}.
    v8f acc = {};
    const float* aSelf = &sH [m0 * HSTRIDE + lrow * HSTRIDE];
    const float* aNei  = &sHn[m0 * HSTRIDE + lrow * HSTRIDE];
    #pragma unroll
    for (int kk = 0; kk < 16; ++kk) {
        const int kb = kk * 4 + khalf * 2;
        v2f a, b;
        a.x = aSelf[kb];
        a.y = aSelf[kb + 1];
        b.x = sWs[(kb    ) * F + c0 + lrow];
        b.y = sWs[(kb + 1) * F + c0 + lrow];
        acc = __builtin_amdgcn_wmma_f32_16x16x4_f32(false, a, false, b,
                                                    (short)0, acc, false, false);
        a.x = aNei[kb];
        a.y = aNei[kb + 1];
        b.x = sWn[(kb    ) * F + c0 + lrow];
        b.y = sWn[(kb + 1) * F + c0 + lrow];
        acc = __builtin_amdgcn_wmma_f32_16x16x4_f32(false, a, false, b,
                                                    (short)0, acc, false, false);
    }

    // C/D layout: VGPR i holds row (m0 + i + 8*khalf), col (c0 + lane%16).
    const float bv = sB[c0 + lrow];
    #pragma unroll
    for (int i = 0; i < 8; ++i) {
        int row = node0 + m0 + i + khalf * 8;
        float v = acc[i] + bv;
        if (do_relu) v = fmaxf(v, 0.0f);
        if (row < n_nodes) out[(long long)row * F + c0 + lrow] = v;
    }
}

extern "C" void kernel_launch(void* const* d_in, const int* in_sizes, int n_in,
                              void* d_out, int out_size, void* d_ws, size_t ws_size,
                              hipStream_t stream) {
    (void)in_sizes; (void)n_in; (void)out_size; (void)ws_size;
    const float* feat = (const float*)d_in[0];
    const int*   src  = (const int*)d_in[1];
    const int*   dst  = (const int*)d_in[2];
    const float* Ws[3] = { (const float*)d_in[3], (const float*)d_in[6], (const float*)d_in[9]  };
    const float* Wn[3] = { (const float*)d_in[4], (const float*)d_in[7], (const float*)d_in[10] };
    const float* bs[3] = { (const float*)d_in[5], (const float*)d_in[8], (const float*)d_in[11] };
    float* outp = (float*)d_out;

    // Workspace layout (floats): deg[N] | msum[N*64] | h1[N*64] | h2[N*64]  (~77.2 MB)
    float* ws   = (float*)d_ws;
    float* deg  = ws;
    float* msum = deg  + (long long)N_NODES;
    float* h1   = msum + (long long)N_NODES * F;
    float* h2   = h1   + (long long)N_NODES * F;

    const int zeroBlocks  = 2048;
    const int degBlocks   = (N_EDGES + 255) / 256;
    const int aggBlocks   = (int)(((long long)N_EDGES * 32 + 255) / 256);
    const int gemmBlocks  = (N_NODES + 31) / 32;

    // Degree (computed fresh every call; deterministic).
    zero_f32<<<zeroBlocks, 256, 0, stream>>>(deg, (long long)N_NODES * (1 + F)); // deg + msum
    sage_degree<<<degBlocks, 256, 0, stream>>>(dst, deg, N_EDGES);

    // Layer 0: features -> h1 (ReLU)
    sage_aggregate<<<aggBlocks, 256, 0, stream>>>(feat, src, dst, msum, N_EDGES);
    sage_gemm<<<gemmBlocks, 256, 0, stream>>>(feat, msum, deg, Ws[0], Wn[0], bs[0], h1, N_NODES, 1);

    // Layer 1: h1 -> h2 (ReLU)
    zero_f32<<<zeroBlocks, 256, 0, stream>>>(msum, (long long)N_NODES * F);
    sage_aggregate<<<aggBlocks, 256, 0, stream>>>(h1, src, dst, msum, N_EDGES);
    sage_gemm<<<gemmBlocks, 256, 0, stream>>>(h1, msum, deg, Ws[1], Wn[1], bs[1], h2, N_NODES, 1);

    // Layer 2: h2 -> out (no ReLU)
    zero_f32<<<zeroBlocks, 256, 0, stream>>>(msum, (long long)N_NODES * F);
    sage_aggregate<<<aggBlocks, 256, 0, stream>>>(h2, src, dst, msum, N_EDGES);
    sage_gemm<<<gemmBlocks, 256, 0, stream>>>(h2, msum, deg, Ws[2], Wn[2], bs[2], outp, N_NODES, 0);
}